// SubgraphEncoder_88673894793797
// MI455X (gfx1250) — compile-verified
//
#include <hip/hip_runtime.h>

#define B_TOTAL  8192
#define E_PER    64
#define P_PER    16
#define D_DIM    128
#define MOTIF_N  64
#define MOTIF_D  128
#define OUT_D    256
#define FUSED_K  256

typedef __attribute__((ext_vector_type(2))) float v2f;
typedef __attribute__((ext_vector_type(8))) float v8f;

// ---------------------------------------------------------------------------
// Kernel 1: gather + mean pool. One wave per batch row; each lane owns one
// float4 (4 of the 128 dims) -> each table-row fetch is one coalesced 512B
// global_load_b128 per wave. Writes semantic into d_out[:, 0:128].
// ---------------------------------------------------------------------------
__global__ __launch_bounds__(256) void gather_mean_kernel(
    const int* __restrict__ entity_idx,     // [B, 64]
    const int* __restrict__ pred_idx,       // [B, 16]
    const float* __restrict__ entity_table, // [200000, 128]
    const float* __restrict__ pred_table,   // [10000, 128]
    float* __restrict__ out)                // [B, 256]; cols 0..127 written
{
    const int wave = threadIdx.x >> 5;
    const int lane = threadIdx.x & 31;
    const int b    = blockIdx.x * 8 + wave;

    const float4* et = (const float4*)entity_table; // row stride = 32 float4
    const float4* pt = (const float4*)pred_table;

    float4 acc_e = {0.f, 0.f, 0.f, 0.f};
    const int* eidx = entity_idx + b * E_PER;
#pragma unroll 4
    for (int e = 0; e < E_PER; ++e) {
        const int r = eidx[e];
        const float4 v = et[r * 32 + lane];
        acc_e.x += v.x; acc_e.y += v.y; acc_e.z += v.z; acc_e.w += v.w;
    }

    float4 acc_p = {0.f, 0.f, 0.f, 0.f};
    const int* pidx = pred_idx + b * P_PER;
#pragma unroll 4
    for (int p = 0; p < P_PER; ++p) {
        const int r = pidx[p];
        const float4 v = pt[r * 32 + lane];
        acc_p.x += v.x; acc_p.y += v.y; acc_p.z += v.z; acc_p.w += v.w;
    }

    const float se = 0.5f / (float)E_PER;   // mean over 64, then *0.5
    const float sp = 0.5f / (float)P_PER;   // mean over 16, then *0.5
    float4 sem;
    sem.x = acc_e.x * se + acc_p.x * sp;
    sem.y = acc_e.y * se + acc_p.y * sp;
    sem.z = acc_e.z * se + acc_p.z * sp;
    sem.w = acc_e.w * se + acc_p.w * sp;

    // out row = 256 floats = 64 float4; semantic occupies first 32 float4.
    ((float4*)out)[b * 64 + lane] = sem;
}

// ---------------------------------------------------------------------------
// WMMA fragment helpers for V_WMMA_F32_16X16X4_F32 (wave32).
// A (16x4, row-major, leading dim lda):
//   lanes 0-15  : M=lane,    VGPR0=K(k0),   VGPR1=K(k0+1)
//   lanes 16-31 : M=lane-16, VGPR0=K(k0+2), VGPR1=K(k0+3)
// B (4x16, row-major K x N): symmetric, N striped across lanes.
// C/D (16x16): VGPR r -> M=r (lanes 0-15) / M=r+8 (lanes 16-31), N=lane&15.
// ---------------------------------------------------------------------------
__device__ __forceinline__ v2f load_a_frag(const float* __restrict__ A, int lda,
                                           int k0, int half, int lidx) {
    const int ka = k0 + half * 2;
    v2f a;
    a.x = A[lidx * lda + ka];
    a.y = A[lidx * lda + ka + 1];
    return a;
}

__device__ __forceinline__ v2f load_b_frag(const float* __restrict__ Bm, int ldb,
                                           int k0, int half, int lidx) {
    const int ka = k0 + half * 2;
    v2f b;
    b.x = Bm[ka * ldb + lidx];
    b.y = Bm[(ka + 1) * ldb + lidx];
    return b;
}

// ---------------------------------------------------------------------------
// Kernel 2: motif MLP. One block = 16 batch rows; 8 waves own the 8 N-tiles
// of MOTIF_D=128. Layer1: K=64 (16 WMMAs). h1 staged in LDS with ld=132
// (132 mod 64 = 4 -> A-fragment lanes hit distinct banks). Layer2: K=128
// (32 WMMAs). Writes m into d_out[:, 128:256].
// ---------------------------------------------------------------------------
__global__ __launch_bounds__(256) void motif_mlp_kernel(
    const float* __restrict__ motif, // [B, 64]
    const float* __restrict__ W1,    // [64, 128]
    const float* __restrict__ b1,    // [128]
    const float* __restrict__ W2,    // [128, 128]
    const float* __restrict__ b2,    // [128]
    float* __restrict__ out)         // [B, 256]; cols 128..255 written
{
    __shared__ float h1[16 * 132];

    const int wave = threadIdx.x >> 5;
    const int lane = threadIdx.x & 31;
    const int half = lane >> 4;
    const int lidx = lane & 15;
    const int row0 = blockIdx.x * 16;
    const int ncol = wave * 16;

    // ---- layer 1: relu(motif @ W1 + b1) -> LDS ----
    const float* A1 = motif + row0 * MOTIF_N;
    v8f acc = {0.f, 0.f, 0.f, 0.f, 0.f, 0.f, 0.f, 0.f};
#pragma unroll
    for (int k = 0; k < MOTIF_N; k += 4) {
        const v2f a = load_a_frag(A1, MOTIF_N, k, half, lidx);
        const v2f b = load_b_frag(W1 + ncol, MOTIF_D, k, half, lidx);
        acc = __builtin_amdgcn_wmma_f32_16x16x4_f32(
            false, a, false, b, (short)0, acc, false, false);
    }
    {
        const float bias = b1[ncol + lidx];
        const int mb = half * 8;
#pragma unroll
        for (int r = 0; r < 8; ++r) {
            float v = acc[r] + bias;
            h1[(mb + r) * 132 + ncol + lidx] = v > 0.f ? v : 0.f;
        }
    }
    __syncthreads();

    // ---- layer 2: relu(h1 @ W2 + b2) -> out[:,128:256] ----
    v8f acc2 = {0.f, 0.f, 0.f, 0.f, 0.f, 0.f, 0.f, 0.f};
#pragma unroll
    for (int k = 0; k < MOTIF_D; k += 4) {
        const v2f a = load_a_frag(h1, 132, k, half, lidx);          // ds_load
        const v2f b = load_b_frag(W2 + ncol, MOTIF_D, k, half, lidx);
        acc2 = __builtin_amdgcn_wmma_f32_16x16x4_f32(
            false, a, false, b, (short)0, acc2, false, false);
    }
    {
        const float bias = b2[ncol + lidx];
        const int mb = half * 8;
#pragma unroll
        for (int r = 0; r < 8; ++r) {
            float v = acc2[r] + bias;
            out[(row0 + mb + r) * OUT_D + 128 + ncol + lidx] = v > 0.f ? v : 0.f;
        }
    }
}

// ---------------------------------------------------------------------------
// Kernel 3: fusion GEMM, in place. fused[B,256] lives in d_out (semantic in
// cols 0:128 from kernel 1, m in cols 128:256 from kernel 2). Each block owns
// 16 rows: stage the 16x256 fused tile into LDS (b128 ds stores, ld=260 for
// bank spread), barrier (so all global reads precede the in-place overwrite),
// then each of 8 waves computes two 16x16 output tiles over K=256.
// ---------------------------------------------------------------------------
__global__ __launch_bounds__(256) void fusion_kernel(
    const float* __restrict__ Wf,  // [256, 256]
    const float* __restrict__ bf,  // [256]
    float* __restrict__ out)       // [B, 256]: fused in, result out (in place)
{
    __shared__ float As[16 * 260];

    const int wave = threadIdx.x >> 5;
    const int lane = threadIdx.x & 31;
    const int half = lane >> 4;
    const int lidx = lane & 15;
    const int row0 = blockIdx.x * 16;
    const int n0   = wave * 32; // two adjacent 16-wide N tiles per wave

    // cooperative stage: 16 rows x 64 float4 = 1024 float4, 4 per thread
    const float4* src = (const float4*)(out + row0 * OUT_D);
#pragma unroll
    for (int i = 0; i < 4; ++i) {
        const int idx = i * 256 + threadIdx.x; // 0..1023
        const int r   = idx >> 6;
        const int c4  = idx & 63;
        *(float4*)&As[r * 260 + c4 * 4] = src[r * 64 + c4];
    }
    __syncthreads();

    v8f acc0 = {0.f, 0.f, 0.f, 0.f, 0.f, 0.f, 0.f, 0.f};
    v8f acc1 = {0.f, 0.f, 0.f, 0.f, 0.f, 0.f, 0.f, 0.f};
#pragma unroll 8
    for (int k = 0; k < FUSED_K; k += 4) {
        const v2f a  = load_a_frag(As, 260, k, half, lidx);
        const v2f bA = load_b_frag(Wf + n0, OUT_D, k, half, lidx);
        const v2f bB = load_b_frag(Wf + n0 + 16, OUT_D, k, half, lidx);
        acc0 = __builtin_amdgcn_wmma_f32_16x16x4_f32(
            false, a, false, bA, (short)0, acc0, false, false);
        acc1 = __builtin_amdgcn_wmma_f32_16x16x4_f32(
            false, a, false, bB, (short)0, acc1, false, false);
    }

    const float bias0 = bf[n0 + lidx];
    const float bias1 = bf[n0 + 16 + lidx];
    const int mb = half * 8;
#pragma unroll
    for (int r = 0; r < 8; ++r) {
        float v0 = acc0[r] + bias0; v0 = v0 > 0.f ? v0 : 0.f;
        float v1 = acc1[r] + bias1; v1 = v1 > 0.f ? v1 : 0.f;
        out[(row0 + mb + r) * OUT_D + n0 + lidx]      = v0;
        out[(row0 + mb + r) * OUT_D + n0 + 16 + lidx] = v1;
    }
}

// ---------------------------------------------------------------------------
extern "C" void kernel_launch(void* const* d_in, const int* in_sizes, int n_in,
                              void* d_out, int out_size, void* d_ws, size_t ws_size,
                              hipStream_t stream) {
    (void)in_sizes; (void)n_in; (void)out_size; (void)d_ws; (void)ws_size;

    const int*   entity_idx   = (const int*)d_in[0];
    const int*   pred_idx     = (const int*)d_in[1];
    const float* motif_counts = (const float*)d_in[2];
    const float* entity_table = (const float*)d_in[3];
    const float* pred_table   = (const float*)d_in[4];
    const float* W1           = (const float*)d_in[5];
    const float* b1           = (const float*)d_in[6];
    const float* W2           = (const float*)d_in[7];
    const float* b2           = (const float*)d_in[8];
    const float* Wf           = (const float*)d_in[9];
    const float* bf           = (const float*)d_in[10];
    float*       out          = (float*)d_out;

    // 1) semantic -> out[:, 0:128]   (memory-bound gather; L2-resident tables)
    gather_mean_kernel<<<B_TOTAL / 8, 256, 0, stream>>>(
        entity_idx, pred_idx, entity_table, pred_table, out);

    // 2) motif MLP -> out[:, 128:256]  (fp32 WMMA)
    motif_mlp_kernel<<<B_TOTAL / 16, 256, 0, stream>>>(
        motif_counts, W1, b1, W2, b2, out);

    // 3) fused @ Wf + bf, relu, in place over out  (fp32 WMMA, K=256)
    fusion_kernel<<<B_TOTAL / 16, 256, 0, stream>>>(Wf, bf, out);
}